// LinearAttention_5763846111248
// MI455X (gfx1250) — compile-verified
//
#include <hip/hip_runtime.h>

// CDNA5 / gfx1250, wave32.
typedef __attribute__((ext_vector_type(2))) float v2f;   // 16x4 (or 4x16) f32 A/B operand: 64 f32 / 32 lanes
typedef __attribute__((ext_vector_type(8))) float v8f;   // 16x16 f32 C/D operand: 256 f32 / 32 lanes

#define B_DIM 8
#define N_DIM 64
#define D_DIM 256
#define K_UPD 3

// ---------------------------------------------------------------------------
// Kernel A: streaming copy M -> out.
// 128 MB in + 128 MB out ~= 256 MB @ 23.3 TB/s => ~11 us floor. b128 per lane.
// ---------------------------------------------------------------------------
__global__ void la_copy_kernel(const float4* __restrict__ src,
                               float4* __restrict__ dst, long n4) {
  long i = (long)blockIdx.x * blockDim.x + threadIdx.x;
  if (i < n4) dst[i] = src[i];
}

// ---------------------------------------------------------------------------
// Kernel B: rank-1 scatter-add via V_WMMA_F32_16X16X4_F32.
// One block per (b, k). Earliest k owning a given row index accumulates all
// duplicates (chained WMMA through C), then out_tile = loaded_tile + sum.
//
// f32 WMMA operand layouts (ISA 7.12.2):
//   A 16x4:  lanes 0-15  -> VGPR0 = K=0 (our k-vector), VGPR1 = K=1
//            lanes 16-31 -> K=2,3            => only lanes 0-15, .x populated
//   B 4x16:  lanes 0-15  -> VGPR0 = row K=0 (our v-vector)
//   C/D 16x16: lane<16: VGPR r = row r; lane>=16: VGPR r = row r+8
// ---------------------------------------------------------------------------
__global__ void la_rank1_wmma_kernel(const float* __restrict__ Mk,
                                     const float* __restrict__ Mv,
                                     const long long* __restrict__ idx,
                                     float* __restrict__ out) {
  const int b = blockIdx.x / K_UPD;
  const int k = blockIdx.x % K_UPD;
  const long long n = idx[b * K_UPD + k];

  // Ownership: the first k with this row index handles all duplicates.
  // Uniform (scalar) decision -> whole block returns or none does.
  for (int kp = 0; kp < k; ++kp)
    if (idx[b * K_UPD + kp] == n) return;

  const int lane   = threadIdx.x & 31;  // wave32
  const int wave   = threadIdx.x >> 5;  // 8 waves per block
  const int lo     = lane & 15;
  const bool loHalf = lane < 16;

  const float* kbase = Mk + ((long)b * N_DIM + n) * D_DIM;
  const float* vbase = Mv + ((long)b * N_DIM + n) * D_DIM;
  float* row_out = out + (((long)b * N_DIM + n) * D_DIM) * D_DIM;

  // 256x256 row-slab = 16x16 grid of 16x16 tiles; 256 tiles over 8 waves.
  for (int t = wave; t < 256; t += 8) {
    const int tm = t >> 4;        // tile row
    const int tn = t & 15;        // tile col

    // Load current out tile into C (stride D_DIM between the 8 row-VGPRs).
    const int mrow0 = tm * 16 + (loHalf ? 0 : 8);
    const int ncol  = tn * 16 + lo;
    float* p = row_out + (long)mrow0 * D_DIM + ncol;
    v8f c;
#pragma unroll
    for (int r = 0; r < 8; ++r) c[r] = p[r * D_DIM];

    // Accumulate every duplicate occurrence of this row index.
    for (int kk = k; kk < K_UPD; ++kk) {
      if (idx[b * K_UPD + kk] != n) continue;   // uniform branch

      v2f a = {0.0f, 0.0f};   // A: only K=0 column nonzero
      v2f bm = {0.0f, 0.0f};  // B: only row K=0 nonzero
      if (loHalf) {
        a.x  = kbase[tm * 16 + lo];  // rows of the outer product
        bm.x = vbase[tn * 16 + lo];  // cols of the outer product
      }
      // D = A x B + C  (EXEC is all-1s here; per-lane guard above only
      // masked the loads, not the WMMA).
      c = __builtin_amdgcn_wmma_f32_16x16x4_f32(
          /*neg_a=*/false, a, /*neg_b=*/false, bm,
          /*c_mod=*/(short)0, c, /*reuse_a=*/false, /*reuse_b=*/false);
    }

#pragma unroll
    for (int r = 0; r < 8; ++r) p[r * D_DIM] = c[r];
  }
}

// ---------------------------------------------------------------------------
// Inputs (setup_inputs order): M (B*N*D*D f32), M_k (B*N*D f32),
// M_v (B*N*D f32), indices_update (B*K int64).
// ---------------------------------------------------------------------------
extern "C" void kernel_launch(void* const* d_in, const int* in_sizes, int n_in,
                              void* d_out, int out_size, void* d_ws, size_t ws_size,
                              hipStream_t stream) {
  (void)n_in; (void)d_ws; (void)ws_size; (void)in_sizes;

  const float*     M   = (const float*)d_in[0];
  const float*     Mk  = (const float*)d_in[1];
  const float*     Mv  = (const float*)d_in[2];
  const long long* idx = (const long long*)d_in[3];  // jnp.int64 per reference
  float*           out = (float*)d_out;

  // Kernel A: bulk copy (out_size = 33,554,432 floats, divisible by 4).
  const long n4 = (long)out_size / 4;
  const int threads = 256;
  const unsigned blocks = (unsigned)((n4 + threads - 1) / threads);
  la_copy_kernel<<<blocks, threads, 0, stream>>>(
      (const float4*)M, (float4*)out, n4);

  // Kernel B: 24 row updates, one block per (b,k), 8 waves each.
  la_rank1_wmma_kernel<<<B_DIM * K_UPD, 256, 0, stream>>>(Mk, Mv, idx, out);
}